// YOLOLoss_38414187495677
// MI455X (gfx1250) — compile-verified
//
#include <hip/hip_runtime.h>
#include <math.h>

// ---------------------------------------------------------------------------
// YOLO loss: sum over cells of coord + BCE(conf) + class SSE, gated on x!=0.
// Bandwidth-bound streaming reduction (192.7 MB in, 4 B out; ~8.3us at 23.3TB/s).
// Wave-cooperative 128-cell tiles staged global->LDS (gfx1250 async DMA path),
// conflict-free LDS reads, deterministic two-pass reduction, WMMA f32 16x16x4
// cross-lane fold in the final kernel.
// ---------------------------------------------------------------------------

#define SLOTS 15
#define WAVES_PER_BLOCK 4
#define BLOCK_THREADS (WAVES_PER_BLOCK * 32)
#define TILE_CELLS 128                       // cells staged per wave-iteration
#define TILE_FLOATS (TILE_CELLS * SLOTS)     // 1920 floats = 7680 B = 480 float4
#define MAX_BLOCKS 2048
#define YEPS 1e-9f

typedef float v2f __attribute__((ext_vector_type(2)));
typedef float v8f __attribute__((ext_vector_type(8)));

// Exact parameter types for __builtin_amdgcn_global_load_async_to_lds_b128
// (from clang's diagnostic): non-const GCC-vector int4 in AS1 / AS3.
typedef int v4i_gcc __attribute__((__vector_size__(16)));
typedef __attribute__((address_space(1))) v4i_gcc* gasync_ptr;
typedef __attribute__((address_space(3))) v4i_gcc* lasync_ptr;

__device__ __forceinline__ float cell_loss(const float* o, const float* t) {
  const float ox = o[0];
  const float d0 = ox - t[0];
  const float dw = sqrtf(o[2]) - sqrtf(t[2]);
  const float dh = sqrtf(o[3]) - sqrtf(t[3]);
  const float coord = d0 * d0 + dw * dw + dh * dh;
  const float oc = o[4], tc = t[4];
  const float conf = -(tc * logf(oc + YEPS) + (1.0f - tc) * logf(1.0f - oc + YEPS));
  float cls = 0.0f;
#pragma unroll
  for (int s = 5; s < 14; ++s) {
    const float d = t[s] - o[s];
    cls += d * d;
  }
  return (ox != 0.0f) ? (coord + conf + cls) : 0.0f;
}

__global__ void __launch_bounds__(BLOCK_THREADS)
yolo_loss_main(const float* __restrict__ out, const float* __restrict__ tgt,
               float* __restrict__ block_sums, long long n_cells) {
  __shared__ __align__(16) float sho[WAVES_PER_BLOCK][TILE_FLOATS];
  __shared__ __align__(16) float sht[WAVES_PER_BLOCK][TILE_FLOATS];
  __shared__ float wred[WAVES_PER_BLOCK];

  const int lane = threadIdx.x & 31;
  const int wave = threadIdx.x >> 5;
  const long long n_tiles = n_cells / TILE_CELLS;
  const long long gwave = (long long)blockIdx.x * WAVES_PER_BLOCK + wave;
  const long long wstride = (long long)gridDim.x * WAVES_PER_BLOCK;

  float acc = 0.0f;

  for (long long tile = gwave; tile < n_tiles; tile += wstride) {
    const long long base = tile * (long long)TILE_FLOATS;  // 16B-aligned elem offset

#if defined(__gfx1250__) && __has_builtin(__builtin_amdgcn_global_load_async_to_lds_b128)
    {
      // Ensure prior-iteration LDS reads fully retired before DMA overwrites.
      asm volatile("s_wait_dscnt 0x0" ::: "memory");
#pragma unroll
      for (int k = 0; k < 15; ++k) {
        const int q = (k * 32 + lane) * 4;  // float index, 16B aligned
        __builtin_amdgcn_global_load_async_to_lds_b128(
            (gasync_ptr)(out + base + q), (lasync_ptr)(&sho[wave][q]), 0, 0);
        __builtin_amdgcn_global_load_async_to_lds_b128(
            (gasync_ptr)(tgt + base + q), (lasync_ptr)(&sht[wave][q]), 0, 0);
      }
#if __has_builtin(__builtin_amdgcn_s_wait_asynccnt)
      __builtin_amdgcn_s_wait_asynccnt(0);
#else
      asm volatile("s_wait_asynccnt 0x0" ::: "memory");
#endif
      asm volatile("" ::: "memory");
    }
#else
    // Synchronous staged copy: coalesced b128 global loads -> b128 LDS stores.
#pragma unroll
    for (int k = 0; k < 15; ++k) {
      const int q = (k * 32 + lane) * 4;
      const float4 a = *(const float4*)(out + base + q);
      const float4 b = *(const float4*)(tgt + base + q);
      *(float4*)(&sho[wave][q]) = a;
      *(float4*)(&sht[wave][q]) = b;
    }
#endif

    // 4 cells per lane; cell stride 15 dwords is coprime with 64 banks ->
    // conflict-free per-slot access pattern.
#pragma unroll
    for (int j = 0; j < 4; ++j) {
      const int c = lane + 32 * j;
      acc += cell_loss(&sho[wave][c * SLOTS], &sht[wave][c * SLOTS]);
    }
  }

  // Tail cells (n_cells not a multiple of TILE_CELLS): direct scalar path.
  {
    const long long tail0 = n_tiles * (long long)TILE_CELLS;
    const long long tid = (long long)blockIdx.x * blockDim.x + threadIdx.x;
    const long long cell = tail0 + tid;
    if (cell < n_cells) {
      float o[SLOTS], t[SLOTS];
      const long long eb = cell * SLOTS;
#pragma unroll
      for (int s = 0; s < SLOTS; ++s) {
        o[s] = out[eb + s];
        t[s] = tgt[eb + s];
      }
      acc += cell_loss(o, t);
    }
  }

  // Wave32 reduction, then block reduction, deterministic order.
  acc += __shfl_xor(acc, 16, 32);
  acc += __shfl_xor(acc, 8, 32);
  acc += __shfl_xor(acc, 4, 32);
  acc += __shfl_xor(acc, 2, 32);
  acc += __shfl_xor(acc, 1, 32);
  if (lane == 0) wred[wave] = acc;
  __syncthreads();
  if (threadIdx.x == 0) {
    float s = 0.0f;
#pragma unroll
    for (int w = 0; w < WAVES_PER_BLOCK; ++w) s += wred[w];
    block_sums[blockIdx.x] = s;
  }
}

// Final deterministic reduction: one wave. Each lane folds block_sums[lane::32]
// in fixed order, then a WMMA f32 16x16x4 with A=ones collapses lane n and
// lane n+16 pairs (D[m][n] = p_n + p_(n+16) regardless of row-mapping detail),
// finished by 4 xor-shuffles over the 16-lane halves.
__global__ void __launch_bounds__(32)
yolo_loss_final(const float* __restrict__ block_sums, int n, float* __restrict__ outv) {
  const int lane = threadIdx.x;
  float g = 0.0f;
  for (int i = lane; i < n; i += 32) g += block_sums[i];

#if defined(__gfx1250__) && __has_builtin(__builtin_amdgcn_wmma_f32_16x16x4_f32)
  v2f a;
  a.x = 1.0f; a.y = 1.0f;           // A (16x4) = all ones under any layout
  v2f b;
  b.x = g;    b.y = 0.0f;           // two of B's four rows carry lane values
  v8f c = {};
  c = __builtin_amdgcn_wmma_f32_16x16x4_f32(
      /*neg_a=*/false, a, /*neg_b=*/false, b,
      /*c_mod=*/(short)0, c, /*reuse_a=*/false, /*reuse_b=*/false);
  float v = c[0];                   // lane L holds p_(L%16) + p_(L%16+16)
  v += __shfl_xor(v, 1, 32);
  v += __shfl_xor(v, 2, 32);
  v += __shfl_xor(v, 4, 32);
  v += __shfl_xor(v, 8, 32);
#else
  float v = g;
  v += __shfl_xor(v, 16, 32);
  v += __shfl_xor(v, 8, 32);
  v += __shfl_xor(v, 4, 32);
  v += __shfl_xor(v, 2, 32);
  v += __shfl_xor(v, 1, 32);
#endif
  if (lane == 0) outv[0] = v;
}

extern "C" void kernel_launch(void* const* d_in, const int* in_sizes, int n_in,
                              void* d_out, int out_size, void* d_ws, size_t ws_size,
                              hipStream_t stream) {
  const float* o = (const float*)d_in[0];
  const float* t = (const float*)d_in[1];
  float* block_sums = (float*)d_ws;

  const long long n_elems = (long long)in_sizes[0];
  const long long n_cells = n_elems / SLOTS;
  const long long n_tiles = n_cells / TILE_CELLS;

  long long want = (n_tiles + WAVES_PER_BLOCK - 1) / WAVES_PER_BLOCK;
  if (want < 1) want = 1;
  if (want > MAX_BLOCKS) want = MAX_BLOCKS;
  const int blocks = (int)want;

  yolo_loss_main<<<blocks, BLOCK_THREADS, 0, stream>>>(o, t, block_sums, n_cells);
  yolo_loss_final<<<1, 32, 0, stream>>>(block_sums, blocks, (float*)d_out);
}